// KSpaceResampling_71433896067140
// MI455X (gfx1250) — compile-verified
//
#include <hip/hip_runtime.h>
#include <hip/hip_bf16.h>
#include <math.h>

// ---------------------------------------------------------------------------
// NUFFT type2 -> Jackson density weights -> type1 adjoint -> abs
// H=W=96, VIEWS=144, SAMPLES=192, M=27648, images = C*B = 8.
// Heavy contractions on V_WMMA_F32_16X16X4_F32 (fp32 matrix core); complex
// GEMM = 4 real WMMA accumulations (2 when A is real). Phases generated with
// __sincosf only at phasor bases; neighbouring tiles/rows advance by complex
// rotation (full-rate FMA) to keep the quarter-rate TRANS pipe off the
// critical path. K (=M) split across blocks with deterministic fixed-order
// partial reductions (no FP atomics).
// ---------------------------------------------------------------------------

#define HH 96
#define WW 96
#define VIEWS 144
#define SAMPLES 192
#define MPTS (VIEWS * SAMPLES)   // 27648
#define NIMG 8                   // C*B, image index i = c*2 + b
#define TWO_PI 6.28318530717958647692f

#define KSPLIT_D 8    // K-chunks for density gridding GEMM
#define KSPLIT_A 16   // K-chunks for adjoint GEMM

// ws layout (floats), total ~12.2 MB
#define WS_KX    0
#define WS_KY    (MPTS)
#define WS_WTS   (2 * MPTS)
#define WS_IMG1  (3 * MPTS)                         // [H][W] cplx
#define WS_KSP   (WS_IMG1 + 2 * HH * WW)            // [NIMG][M] cplx
#define WS_PARTD (WS_KSP + 2 * NIMG * MPTS)         // [KSPLIT_D][H][W] cplx
#define WS_PARTA (WS_PARTD + KSPLIT_D * 2 * HH * WW) // [KSPLIT_A][NIMG][H][W] cplx

typedef __attribute__((ext_vector_type(2))) float v2f;
typedef __attribute__((ext_vector_type(8))) float v8f;

__device__ __forceinline__ v8f wmma4(v2f a, v2f b, v8f c) {
  // D = A(16x4) * B(4x16) + C ; f32 path (neg_a/neg_b must be 0 for f32)
  return __builtin_amdgcn_wmma_f32_16x16x4_f32(false, a, false, b, (short)0, c,
                                               false, false);
}

// ---------------------------------------------------------------------------
// Kernel A: radial trajectory (linear ordering, full angle range)
// ---------------------------------------------------------------------------
__global__ void k_traj(float* __restrict__ ws) {
  int m = blockIdx.x * blockDim.x + threadIdx.x;
  if (m >= MPTS) return;
  int view = m / SAMPLES;
  int s = m - view * SAMPLES;
  float r = ((float)s / (float)SAMPLES - 0.5f) * TWO_PI;
  float theta = TWO_PI * (float)view / (float)VIEWS;
  float st, ct;
  __sincosf(theta, &st, &ct);
  ws[WS_KX + m] = ct * r;
  ws[WS_KY + m] = st * r;
}

// ---------------------------------------------------------------------------
// Kernel B: img1 = type1(ones) partials: img1[h,w] = sum_m e^{+i(kx xh + ky yw)}
// Complex WMMA GEMM, one wave per (16x16 tile, K-chunk).
// ---------------------------------------------------------------------------
__global__ __launch_bounds__(32) void k_density_img1(float* __restrict__ ws) {
  const int kchunk = blockIdx.x / 36;
  const int tile = blockIdx.x % 36;
  const int ht = tile % (HH / 16);
  const int wt = tile / (HH / 16);
  const int lane = threadIdx.x;
  const int half = lane >> 4;
  const int l = lane & 15;

  const float xh = (float)(ht * 16 + l) - 48.0f;
  const float yw = (float)(wt * 16 + l) - 48.0f;

  const int mbeg = kchunk * (MPTS / KSPLIT_D);
  const int mend = mbeg + (MPTS / KSPLIT_D);

  v8f cre = {};
  v8f cim = {};

  for (int ms = mbeg; ms < mend; ms += 4) {
    int m0 = ms + 2 * half;
    float kx0 = ws[WS_KX + m0], kx1 = ws[WS_KX + m0 + 1];
    float ky0 = ws[WS_KY + m0], ky1 = ws[WS_KY + m0 + 1];
    float s0, c0, s1, c1;
    __sincosf(kx0 * xh, &s0, &c0);
    __sincosf(kx1 * xh, &s1, &c1);
    v2f are = {c0, c1}, aim = {s0, s1};
    __sincosf(ky0 * yw, &s0, &c0);
    __sincosf(ky1 * yw, &s1, &c1);
    v2f bre = {c0, c1}, bim = {s0, s1};
    v2f naim = {-aim.x, -aim.y};
    cre = wmma4(are, bre, cre);
    cre = wmma4(naim, bim, cre);
    cim = wmma4(are, bim, cim);
    cim = wmma4(aim, bre, cim);
  }

#pragma unroll
  for (int r = 0; r < 8; ++r) {
    int h = ht * 16 + r + 8 * half;
    int w = wt * 16 + l;
    ws[WS_PARTD + 2 * ((kchunk * HH + h) * WW + w) + 0] = cre[r];
    ws[WS_PARTD + 2 * ((kchunk * HH + h) * WW + w) + 1] = cim[r];
  }
}

// Fixed-order reduction of density partials -> img1 (deterministic).
__global__ void k_img1_reduce(float* __restrict__ ws) {
  int t = blockIdx.x * blockDim.x + threadIdx.x;
  if (t >= HH * WW) return;
  float re = 0.0f, im = 0.0f;
  for (int k = 0; k < KSPLIT_D; ++k) {
    re += ws[WS_PARTD + 2 * (k * HH * WW + t) + 0];
    im += ws[WS_PARTD + 2 * (k * HH * WW + t) + 1];
  }
  ws[WS_IMG1 + 2 * t + 0] = re;
  ws[WS_IMG1 + 2 * t + 1] = im;
}

// ---------------------------------------------------------------------------
// Kernel C: Jackson density + weights.
// dens[m] = (1/9216) * sum_{h,w} [re1*cos(psi) + im1*sin(psi)], psi = kx xh + ky yw
// Rotation recurrence along w: ~1 sincos per h.
// ---------------------------------------------------------------------------
__global__ void k_weights(float* __restrict__ ws) {
  int m = blockIdx.x * blockDim.x + threadIdx.x;
  if (m >= MPTS) return;
  float kxm = ws[WS_KX + m];
  float kym = ws[WS_KY + m];
  float sk, ck;
  __sincosf(kym, &sk, &ck);  // per-w rotation e^{-i ky}
  float dens = 0.0f;
  for (int h = 0; h < HH; ++h) {
    float xh = (float)h - 48.0f;
    float psi0 = kxm * xh - 48.0f * kym;  // w = 0 -> yw = -48
    float s, c;
    __sincosf(psi0, &s, &c);  // e^{-i psi} = c - i s
    const float* row = ws + WS_IMG1 + 2 * h * WW;
    for (int w = 0; w < WW; ++w) {
      dens = fmaf(row[2 * w], c, dens);
      dens = fmaf(row[2 * w + 1], s, dens);
      float cn = c * ck - s * sk;
      float sn = s * ck + c * sk;
      c = cn;
      s = sn;
    }
  }
  dens *= (1.0f / (float)(HH * WW));
  ws[WS_WTS + m] = (dens != 0.0f) ? (1.0f / dens) : 0.0f;
}

// ---------------------------------------------------------------------------
// Kernel D: forward type2 + weighting.
// Grid: M/16 blocks of 8 waves; wave i handles image i (i = c*2 + b).
// GEMM over w (K=96): T[h,m] = sum_w X[h,w]*Ey[m,w]  (A real -> 2 WMMA/step)
// Epilogue: F[m] = sum_h Ex[m,h]*T[h,m] via phase rotation; ksp = F*wts[m].
// ---------------------------------------------------------------------------
__global__ __launch_bounds__(256) void k_forward(const float* __restrict__ x,
                                                 float* __restrict__ ws) {
  const int mtile = blockIdx.x;
  const int i = threadIdx.x >> 5;
  const int lane = threadIdx.x & 31;
  const int half = lane >> 4;
  const int l = lane & 15;
  const int c_ = i >> 1;
  const int b_ = i & 1;

  const int mn = mtile * 16 + l;
  const float kxm = ws[WS_KX + mn];
  const float kym = ws[WS_KY + mn];
  const float wtm = ws[WS_WTS + mn];

  v8f tre[6], tim[6];
#pragma unroll
  for (int t = 0; t < 6; ++t) { tre[t] = (v8f){}; tim[t] = (v8f){}; }

  for (int w0 = 0; w0 < WW; w0 += 4) {
    int wA = w0 + 2 * half;
    float ywA = (float)wA - 48.0f;
    float s0, c0, s1, c1;
    __sincosf(kym * ywA, &s0, &c0);            // Ey = cos - i sin
    __sincosf(kym * (ywA + 1.0f), &s1, &c1);
    v2f bre = {c0, c1};
    v2f bim = {-s0, -s1};
#pragma unroll
    for (int t = 0; t < 6; ++t) {
      int h = t * 16 + l;
      v2f a;
      a.x = x[((b_ * HH + h) * WW + wA) * 4 + c_];
      a.y = x[((b_ * HH + h) * WW + wA + 1) * 4 + c_];
      tre[t] = wmma4(a, bre, tre[t]);
      tim[t] = wmma4(a, bim, tim[t]);
    }
  }

  // reduce over h with Ex[m,h] = cos(kx xh) - i sin(kx xh); advance by
  // rotation e^{i kx} per row instead of per-row sincos.
  float rs, rc;
  __sincosf(kxm, &rs, &rc);
  float fr = 0.0f, fi = 0.0f;
#pragma unroll
  for (int t = 0; t < 6; ++t) {
    float xh0 = (float)(t * 16 + 8 * half) - 48.0f;
    float es, ec;
    __sincosf(kxm * xh0, &es, &ec);
#pragma unroll
    for (int r = 0; r < 8; ++r) {
      float trv = tre[t][r], tiv = tim[t][r];
      fr += trv * ec + tiv * es;
      fi += tiv * ec - trv * es;
      float ecn = ec * rc - es * rs;  // angle += kxm
      float esn = es * rc + ec * rs;
      ec = ecn;
      es = esn;
    }
  }
  fr += __shfl_xor(fr, 16);
  fi += __shfl_xor(fi, 16);
  if (half == 0) {
    ws[WS_KSP + 2 * (i * MPTS + mn) + 0] = fr * wtm;
    ws[WS_KSP + 2 * (i * MPTS + mn) + 1] = fi * wtm;
  }
}

// ---------------------------------------------------------------------------
// Kernel E: adjoint type1 partials.
// Grid: KSPLIT_A * NIMG * 6 single-wave blocks = (K-chunk, image, row tile).
// Complex GEMM over K = M-chunk:
//   A[h,k] = ksp[i,m] * e^{+i kx xh} ;  B[k,w] = e^{+i ky yw}
// B base phasor per step, 6 column tiles advanced by rotation e^{+i ky*16}.
// ---------------------------------------------------------------------------
__global__ __launch_bounds__(32) void k_adjoint(float* __restrict__ ws) {
  const int kchunk = blockIdx.x / (NIMG * 6);
  const int rem = blockIdx.x % (NIMG * 6);
  const int i = rem / 6;
  const int rt = rem % 6;
  const int lane = threadIdx.x;
  const int half = lane >> 4;
  const int l = lane & 15;

  const float xh = (float)(rt * 16 + l) - 48.0f;
  const float ywl = (float)l - 48.0f;  // column coordinate of tile ct=0

  const int mbeg = kchunk * (MPTS / KSPLIT_A);
  const int mend = mbeg + (MPTS / KSPLIT_A);

  v8f cre[6], cim[6];
#pragma unroll
  for (int t = 0; t < 6; ++t) { cre[t] = (v8f){}; cim[t] = (v8f){}; }

  for (int ms = mbeg; ms < mend; ms += 4) {
    int m0 = ms + 2 * half;
    float kx0 = ws[WS_KX + m0], kx1 = ws[WS_KX + m0 + 1];
    float ky0 = ws[WS_KY + m0], ky1 = ws[WS_KY + m0 + 1];
    float kr0 = ws[WS_KSP + 2 * (i * MPTS + m0) + 0];
    float ki0 = ws[WS_KSP + 2 * (i * MPTS + m0) + 1];
    float kr1 = ws[WS_KSP + 2 * (i * MPTS + m0) + 2];
    float ki1 = ws[WS_KSP + 2 * (i * MPTS + m0) + 3];
    float s0, c0, s1, c1;
    __sincosf(kx0 * xh, &s0, &c0);  // conj(Ex) = cos + i sin
    __sincosf(kx1 * xh, &s1, &c1);
    v2f are = {kr0 * c0 - ki0 * s0, kr1 * c1 - ki1 * s1};
    v2f aim = {kr0 * s0 + ki0 * c0, kr1 * s1 + ki1 * c1};
    v2f naim = {-aim.x, -aim.y};

    // B base phasors at ct=0 and per-16-column rotation constants
    float bs0, bc0, bs1, bc1, rs0, rc0, rs1, rc1;
    __sincosf(ky0 * ywl, &bs0, &bc0);  // conj(Ey) = cos + i sin
    __sincosf(ky1 * ywl, &bs1, &bc1);
    __sincosf(ky0 * 16.0f, &rs0, &rc0);
    __sincosf(ky1 * 16.0f, &rs1, &rc1);

#pragma unroll
    for (int ct = 0; ct < 6; ++ct) {
      v2f bre = {bc0, bc1};
      v2f bim = {bs0, bs1};
      cre[ct] = wmma4(are, bre, cre[ct]);
      cre[ct] = wmma4(naim, bim, cre[ct]);
      cim[ct] = wmma4(are, bim, cim[ct]);
      cim[ct] = wmma4(aim, bre, cim[ct]);
      // advance yw by +16: multiply phasor by e^{+i ky*16}
      float nc0 = bc0 * rc0 - bs0 * rs0, ns0 = bs0 * rc0 + bc0 * rs0;
      float nc1 = bc1 * rc1 - bs1 * rs1, ns1 = bs1 * rc1 + bc1 * rs1;
      bc0 = nc0; bs0 = ns0; bc1 = nc1; bs1 = ns1;
    }
  }

#pragma unroll
  for (int ct = 0; ct < 6; ++ct) {
#pragma unroll
    for (int r = 0; r < 8; ++r) {
      int h = rt * 16 + r + 8 * half;
      int w = ct * 16 + l;
      long idx = 2 * (((long)(kchunk * NIMG + i) * HH + h) * WW + w);
      ws[WS_PARTA + idx + 0] = cre[ct][r];
      ws[WS_PARTA + idx + 1] = cim[ct][r];
    }
  }
}

// ---------------------------------------------------------------------------
// Kernel F: fixed-order K-chunk reduction + |.|/(H*W) + [B,H,W,C] layout.
// ---------------------------------------------------------------------------
__global__ void k_abs(const float* __restrict__ ws, float* __restrict__ out) {
  int t = blockIdx.x * blockDim.x + threadIdx.x;
  if (t >= NIMG * HH * WW) return;
  int i = t / (HH * WW);
  int hw = t % (HH * WW);
  int h = hw / WW;
  int w = hw % WW;
  int c_ = i >> 1;
  int b_ = i & 1;
  float re = 0.0f, im = 0.0f;
  for (int k = 0; k < KSPLIT_A; ++k) {
    long idx = 2 * (((long)(k * NIMG + i) * HH + h) * WW + w);
    re += ws[WS_PARTA + idx + 0];
    im += ws[WS_PARTA + idx + 1];
  }
  const float inv = 1.0f / (float)(HH * WW);
  re *= inv;
  im *= inv;
  out[((b_ * HH + h) * WW + w) * 4 + c_] = sqrtf(re * re + im * im);
}

// ---------------------------------------------------------------------------
extern "C" void kernel_launch(void* const* d_in, const int* in_sizes, int n_in,
                              void* d_out, int out_size, void* d_ws,
                              size_t ws_size, hipStream_t stream) {
  const float* x = (const float*)d_in[0];  // [2,96,96,4] fp32
  float* out = (float*)d_out;              // [2,96,96,4] fp32
  float* ws = (float*)d_ws;                // ~12.2 MB used

  k_traj<<<(MPTS + 255) / 256, 256, 0, stream>>>(ws);
  k_density_img1<<<36 * KSPLIT_D, 32, 0, stream>>>(ws);
  k_img1_reduce<<<(HH * WW + 255) / 256, 256, 0, stream>>>(ws);
  k_weights<<<(MPTS + 255) / 256, 256, 0, stream>>>(ws);
  k_forward<<<MPTS / 16, 256, 0, stream>>>(x, ws);
  k_adjoint<<<KSPLIT_A * NIMG * 6, 32, 0, stream>>>(ws);
  k_abs<<<(NIMG * HH * WW + 255) / 256, 256, 0, stream>>>(ws, out);
}